// GapDecoder_7275674599529
// MI455X (gfx1250) — compile-verified
//
#include <hip/hip_runtime.h>

// MI455X / gfx1250, wave32.
// Problem: proj = relu(X@W1+b1)@w2  (N=8192, D=512, H=32), out[i,j]=proj[i]+proj[j]+b2.
// Cost model: 256MB output stores @ 23.3TB/s ~= 11us dominates; GEMM (268 MFLOP) is noise.
// Kernel 1: f32 WMMA (V_WMMA_F32_16X16X4_F32) GEMM -> proj (b2/2 folded in), into d_ws.
// Kernel 2: pure streaming outer-sum with 128-bit non-temporal stores.

typedef __attribute__((ext_vector_type(2))) float v2f;
typedef __attribute__((ext_vector_type(4))) float v4f;
typedef __attribute__((ext_vector_type(8))) float v8f;

#define N_SITES 8192
#define DIM 512
#define HID 32

// ---------------- Kernel 1: proj = relu(X@W1+b1)@w2 + b2/2 -------------------
// One wave per 16-row tile. 512 tiles -> 64 blocks x 8 waves.
// A fragment (16x4 f32, 2 VGPRs): lanes 0-15 hold M=0..15 with {K=0,K=1},
// lanes 16-31 hold M=0..15 with {K=2,K=3}.  B (4x16) mirrors: v0={K=0|K=2},
// v1={K=1|K=3} across lane halves, N = lane&15.  C/D: VGPR r = rows r / r+8.
__global__ __launch_bounds__(256) void proj_wmma_kernel(
    const float* __restrict__ x,    // [N, D]
    const float* __restrict__ W1,   // [D, H]
    const float* __restrict__ bias1,// [H]
    const float* __restrict__ w2,   // [H]
    const float* __restrict__ bias2,// [1]
    float* __restrict__ proj)       // [N] (d_ws)
{
    const int lane = threadIdx.x & 31;
    const int wave = threadIdx.x >> 5;
    const int rowTile = blockIdx.x * 8 + wave;   // 0..511
    const int rowBase = rowTile * 16;
    const int hi16 = lane >> 4;                  // which lane half
    const int l16  = lane & 15;

    // per-lane A row pointer; +2*hi16 selects K={0,1} vs K={2,3} sub-pair
    const float* xrow = x + (size_t)(rowBase + l16) * DIM + 2 * hi16;
    const int col0 = l16;          // H-tile 0 column (0..15)
    const int col1 = 16 + l16;     // H-tile 1 column (16..31)

    v8f acc0 = {};
    v8f acc1 = {};

    for (int k = 0; k < DIM; k += 4) {
        // A fragment: contiguous float2 (8B aligned: row stride 2KB, offset 0/8)
        v2f a = *(const v2f*)(xrow + k);
        // B fragments: W1 rows (k+2*hi16) and (k+2*hi16+1), column = col0/col1
        const float* Wk = W1 + (size_t)(k + 2 * hi16) * HID;
        v2f b0, b1f;
        b0.x  = Wk[col0];
        b0.y  = Wk[HID + col0];
        b1f.x = Wk[col1];
        b1f.y = Wk[HID + col1];
        // D = A*B + C  (f32 exact, no precision loss vs reference)
        acc0 = __builtin_amdgcn_wmma_f32_16x16x4_f32(
                   false, a, false, b0,  (short)0, acc0, false, false);
        acc1 = __builtin_amdgcn_wmma_f32_16x16x4_f32(
                   false, a, false, b1f, (short)0, acc1, false, false);
    }

    // Epilogue: relu(S+b1)*w2, reduce over H (columns = lanes within each half).
    const float hb2 = 0.5f * bias2[0];   // gaps adds proj[i]+proj[j], so split b2
    const float bb0 = bias1[col0];
    const float bb1 = bias1[col1];
    const float ww0 = w2[col0];
    const float ww1 = w2[col1];

#pragma unroll
    for (int r = 0; r < 8; ++r) {
        float s0 = acc0[r] + bb0; s0 = s0 > 0.0f ? s0 : 0.0f;
        float s1 = acc1[r] + bb1; s1 = s1 > 0.0f ? s1 : 0.0f;
        float v = s0 * ww0 + s1 * ww1;
        // reduce 16 lanes within each half (xor masks stay inside a 16-group)
        v += __shfl_xor(v, 1, 32);
        v += __shfl_xor(v, 2, 32);
        v += __shfl_xor(v, 4, 32);
        v += __shfl_xor(v, 8, 32);
        if (l16 == 0)
            proj[rowBase + 8 * hi16 + r] = v + hb2;  // lanes 0 & 16 write rows r / r+8
    }
}

// ---------------- Kernel 2: out[i][j] = proj[i] + proj[j] --------------------
// Store-bandwidth bound (256MB > 192MB L2 -> non-temporal streaming stores).
// grid = (2, 8192): each block covers one row x 4096 floats; each thread emits
// 4 x b128 NT stores. proj (32KB) stays L2/WGP$-resident.
__global__ __launch_bounds__(256) void gaps_kernel(
    const float* __restrict__ proj, float* __restrict__ out)
{
    const int i = blockIdx.y;
    const float pi = proj[i];                       // uniform per block
    const v4f* p4 = (const v4f*)proj;
    v4f* o = (v4f*)out + (size_t)i * (N_SITES / 4);

    int c4 = blockIdx.x * 1024 + threadIdx.x;       // float4 index within row
#pragma unroll
    for (int t = 0; t < 4; ++t) {
        v4f pj = p4[c4];
        v4f r = pj + pi;                            // splat add
        __builtin_nontemporal_store(r, o + c4);     // b128 NT store
        c4 += 256;
    }
}

extern "C" void kernel_launch(void* const* d_in, const int* in_sizes, int n_in,
                              void* d_out, int out_size, void* d_ws, size_t ws_size,
                              hipStream_t stream) {
    const float* x     = (const float*)d_in[0];  // [8192,512]
    const float* W1    = (const float*)d_in[1];  // [512,32]
    const float* bias1 = (const float*)d_in[2];  // [32]
    const float* w2    = (const float*)d_in[3];  // [32]
    const float* bias2 = (const float*)d_in[4];  // [1]
    float* proj = (float*)d_ws;                  // 8192 floats = 32KB scratch
    float* out  = (float*)d_out;                 // [8192,8192]

    proj_wmma_kernel<<<dim3(64), dim3(256), 0, stream>>>(x, W1, bias1, w2, bias2, proj);
    gaps_kernel<<<dim3(2, N_SITES), dim3(256), 0, stream>>>(proj, out);
}